// RSSM_45758581571601
// MI455X (gfx1250) — compile-verified
//
#include <hip/hip_runtime.h>

// ---------------------------------------------------------------------------
// RSSM GRUCell + prior heads, fused, FP32 WMMA (gfx1250 / MI455X).
//
// Transposed formulation: per 16-batch tile, D(16 units x 16 batch) =
//   A(weights, 16xK) x B([x|h]^T, Kx16) + C(bias), chained over K in K=4 steps
//   with V_WMMA_F32_16X16X4_F32.
//
// Assumed f32 WMMA register layouts (per CDNA5 ISA striping patterns):
//   A 16x4  : value(M,K) -> lane = M + 16*(K>=2), vgpr = K%2
//   B 4x16  : value(K,N) -> lane = N + 16*(K>=2), vgpr = K%2
//   C/D 16x16: value(M,N) -> lane = N + 16*(M/8), vgpr = M%8
// ---------------------------------------------------------------------------

typedef float v2f __attribute__((ext_vector_type(2)));
typedef float v8f __attribute__((ext_vector_type(8)));

#define WAVE 32
#define ROWW 20                   // LDS row stride in words (80B: 16B aligned, bank-spread)
#define TILES_PER_WAVE 8
#define WAVES_PER_BLOCK 8
#define TILES_PER_BLOCK (TILES_PER_WAVE * WAVES_PER_BLOCK)

__device__ __forceinline__ float fast_exp(float x) {
  return __builtin_amdgcn_exp2f(x * 1.4426950408889634f);   // v_exp_f32
}
__device__ __forceinline__ float fast_rcp(float x) {
  return __builtin_amdgcn_rcpf(x);                          // v_rcp_f32
}
__device__ __forceinline__ float fast_sigmoid(float x) {
  return fast_rcp(1.0f + fast_exp(-x));
}
__device__ __forceinline__ float fast_tanh(float x) {
  // tanh(x) = 1 - 2/(e^{2x}+1); saturates correctly for |x| large.
  return 1.0f - 2.0f * fast_rcp(fast_exp(2.0f * x) + 1.0f);
}

__device__ __forceinline__ v8f wmma4(v2f a, v2f b, v8f c) {
  // (neg_a, A, neg_b, B, c_mod, C, reuse_a, reuse_b)
  return __builtin_amdgcn_wmma_f32_16x16x4_f32(false, a, false, b, (short)0, c,
                                               false, false);
}

extern "C" __global__ void __launch_bounds__(256)
rssm_gru_wmma(const float* __restrict__ prev_z, const float* __restrict__ action,
              const float* __restrict__ prev_h,
              const float* __restrict__ W_ih, const float* __restrict__ W_hh,
              const float* __restrict__ b_ih, const float* __restrict__ b_hh,
              const float* __restrict__ W_mu, const float* __restrict__ b_mu,
              const float* __restrict__ W_lv, const float* __restrict__ b_lv,
              float* __restrict__ out_mu, float* __restrict__ out_lv,
              float* __restrict__ out_h, int ntiles)
{
  __shared__ float lds[WAVES_PER_BLOCK * 2 * 16 * ROWW];   // 20 KB / block
  const int tid  = threadIdx.x;
  const int wave = tid / WAVE;
  const int lane = tid % WAVE;
  const int rb   = lane & 15;     // batch row (B/D column) / weight row (A M)
  const int hi   = lane >> 4;     // lane half selects K pair / M+8 half

  float* sIn  = &lds[wave * 2 * 16 * ROWW];  // prev_h staging (wave-private)
  float* sOut = sIn + 16 * ROWW;             // h staging (wave-private)

  // ---- weights -> WMMA A-layout registers (once per wave) ----------------
  // gate order in torch GRUCell: r (rows 0..15), z (16..31), n (32..47)
  v2f WihA[3], WhhA[3][4], WhdA[4];
  #pragma unroll
  for (int g = 0; g < 3; ++g) {
    const int row = g * 16 + rb;
    v2f a; a[0] = W_ih[row * 4 + 2 * hi + 0];
           a[1] = W_ih[row * 4 + 2 * hi + 1];
    WihA[g] = a;
    #pragma unroll
    for (int q = 0; q < 4; ++q) {
      v2f w; w[0] = W_hh[row * 16 + 4 * q + 2 * hi + 0];
             w[1] = W_hh[row * 16 + 4 * q + 2 * hi + 1];
      WhhA[g][q] = w;
    }
  }
  // heads: rows 0..2 = W_mu, rows 3..5 = W_lv, rows 6..15 = 0 (never stored)
  #pragma unroll
  for (int q = 0; q < 4; ++q) {
    v2f w;
    #pragma unroll
    for (int j = 0; j < 2; ++j) {
      const int col = 4 * q + 2 * hi + j;
      float val = 0.0f;
      if (rb < 3)      val = W_mu[rb * 16 + col];
      else if (rb < 6) val = W_lv[(rb - 3) * 16 + col];
      w[j] = val;
    }
    WhdA[q] = w;
  }

  // ---- bias C-inits: this lane's D rows are M = v + 8*hi -----------------
  float bcR[8], bcZ[8], bcNi[8], bcNh[8], bcH[8];
  #pragma unroll
  for (int v = 0; v < 8; ++v) {
    const int up = v + 8 * hi;
    bcR[v]  = b_ih[ 0 + up] + b_hh[ 0 + up];
    bcZ[v]  = b_ih[16 + up] + b_hh[16 + up];
    bcNi[v] = b_ih[32 + up];                 // i_n bias (kept separate: r scales h_n only)
    bcNh[v] = b_hh[32 + up];
    bcH[v]  = (up < 3) ? b_mu[up] : ((up < 6) ? b_lv[up - 3] : 0.0f);
  }

  for (int i = 0; i < TILES_PER_WAVE; ++i) {
    const int t = blockIdx.x * TILES_PER_BLOCK + i * WAVES_PER_BLOCK + wave;
    if (t >= ntiles) break;
    const int base = t * 16;

    // prefetch next tile's prev_h (global_prefetch_b8); 64B stride per lane
    {
      const int tn = t + WAVES_PER_BLOCK;
      if (tn < ntiles)
        __builtin_prefetch(prev_h + (size_t)tn * 256 + lane * 16, 0, 1);
    }

    // ---- stage prev_h tile (16x16 f32, contiguous 1KB) into padded LDS ---
    const float4* gh4 = reinterpret_cast<const float4*>(prev_h + (size_t)base * 16);
    #pragma unroll
    for (int j = 0; j < 2; ++j) {
      const int e = j * 32 + lane;           // float4 index in tile (0..63)
      const float4 tv = gh4[e];
      const int r = e >> 2, c = (e & 3) * 4;
      *reinterpret_cast<float4*>(&sIn[r * ROWW + c]) = tv;
    }
    // DS is in-order per wave; force completion + compiler ordering.
    asm volatile("s_wait_dscnt 0" ::: "memory");

    // H^T as 4x16 B-slices (WMMA chain) and C/D-layout prev_h (h update)
    v2f HbV[4];
    #pragma unroll
    for (int q = 0; q < 4; ++q) {
      v2f h2; h2[0] = sIn[rb * ROWW + 4 * q + 2 * hi + 0];
              h2[1] = sIn[rb * ROWW + 4 * q + 2 * hi + 1];
      HbV[q] = h2;
    }
    float Hcd[8];
    #pragma unroll
    for (int v = 0; v < 8; ++v)
      Hcd[v] = sIn[rb * ROWW + v + 8 * hi];

    // x^T = [prev_z | action]^T as a 4x16 B tile.
    // Branch-free: both lane halves issue two b32 loads from selected
    // addresses (v_cndmask on the pointer), avoiding exec-mask diamonds.
    const int b = base + rb;
    const float* pz = prev_z + (size_t)b * 3;
    const float* p0 = pz + 2 * hi;                    // z0 (lo half) / z2 (hi half)
    const float* p1 = hi ? (action + b) : (pz + 1);   // z1 (lo half) / action (hi half)
    v2f Xb; Xb[0] = *p0; Xb[1] = *p1;

    // ---- gate pre-activations via WMMA (D = 16 units x 16 batch) ---------
    v8f Dr, Dz, Dni, Dnh;
    #pragma unroll
    for (int v = 0; v < 8; ++v) { Dr[v] = bcR[v]; Dz[v] = bcZ[v]; Dni[v] = bcNi[v]; Dnh[v] = bcNh[v]; }
    Dr  = wmma4(WihA[0], Xb, Dr);
    Dz  = wmma4(WihA[1], Xb, Dz);
    Dni = wmma4(WihA[2], Xb, Dni);
    #pragma unroll
    for (int q = 0; q < 4; ++q) {
      Dr  = wmma4(WhhA[0][q], HbV[q], Dr);
      Dz  = wmma4(WhhA[1][q], HbV[q], Dz);
      Dnh = wmma4(WhhA[2][q], HbV[q], Dnh);
    }

    // ---- GRU gate math (lane = batch rb, units v + 8*hi) -----------------
    float hreg[8];
    #pragma unroll
    for (int v = 0; v < 8; ++v) {
      const float r = fast_sigmoid(Dr[v]);
      const float z = fast_sigmoid(Dz[v]);
      const float n = fast_tanh(fmaf(r, Dnh[v], Dni[v]));
      hreg[v] = fmaf(z, Hcd[v] - n, n);      // (1-z)*n + z*prev_h
    }

    // ---- stage h (row=batch, col=unit) for head GEMM + coalesced store ---
    #pragma unroll
    for (int v = 0; v < 8; ++v)
      sOut[rb * ROWW + v + 8 * hi] = hreg[v];
    asm volatile("s_wait_dscnt 0" ::: "memory");

    // heads: D = [W_mu;W_lv] (padded 16x16) x h^T + b, chained K=16
    v8f Dh;
    #pragma unroll
    for (int v = 0; v < 8; ++v) Dh[v] = bcH[v];
    #pragma unroll
    for (int q = 0; q < 4; ++q) {
      v2f hb; hb[0] = sOut[rb * ROWW + 4 * q + 2 * hi + 0];
              hb[1] = sOut[rb * ROWW + 4 * q + 2 * hi + 1];
      Dh = wmma4(WhdA[q], hb, Dh);
    }

    // coalesced h store (b128)
    float4* oh4 = reinterpret_cast<float4*>(out_h + (size_t)base * 16);
    #pragma unroll
    for (int j = 0; j < 2; ++j) {
      const int e = j * 32 + lane;
      const int r = e >> 2, c = (e & 3) * 4;
      oh4[e] = *reinterpret_cast<const float4*>(&sOut[r * ROWW + c]);
    }

    // mu / logvar: lanes 0..15 hold head rows 0..7 (rows 0..5 valid)
    if (hi == 0) {
      #pragma unroll
      for (int k = 0; k < 3; ++k)
        out_mu[(size_t)b * 3 + k] = Dh[k];
      #pragma unroll
      for (int k = 0; k < 3; ++k) {
        float lv = Dh[3 + k];
        lv = fminf(fmaxf(lv, -5.0f), 5.0f);
        out_lv[(size_t)b * 3 + k] = lv;
      }
    }
  }
}

extern "C" void kernel_launch(void* const* d_in, const int* in_sizes, int n_in,
                              void* d_out, int out_size, void* d_ws, size_t ws_size,
                              hipStream_t stream) {
  const float* prev_z = (const float*)d_in[0];
  const float* action = (const float*)d_in[1];
  const float* prev_h = (const float*)d_in[2];
  const float* W_ih   = (const float*)d_in[3];
  const float* W_hh   = (const float*)d_in[4];
  const float* b_ih   = (const float*)d_in[5];
  const float* b_hh   = (const float*)d_in[6];
  const float* W_mu   = (const float*)d_in[7];
  const float* b_mu   = (const float*)d_in[8];
  const float* W_lv   = (const float*)d_in[9];
  const float* b_lv   = (const float*)d_in[10];

  const int B = in_sizes[1];                 // action is [B, 1]
  float* out    = (float*)d_out;             // mu | logvar | h, concatenated
  float* out_mu = out;
  float* out_lv = out + (size_t)B * 3;
  float* out_h  = out + (size_t)B * 6;

  const int ntiles = B / 16;                 // B = 1048576 -> 65536 tiles
  const int grid   = (ntiles + TILES_PER_BLOCK - 1) / TILES_PER_BLOCK;

  rssm_gru_wmma<<<grid, 256, 0, stream>>>(prev_z, action, prev_h, W_ih, W_hh,
                                          b_ih, b_hh, W_mu, b_mu, W_lv, b_lv,
                                          out_mu, out_lv, out_h, ntiles);
}